// ReRaLSTM_57277683859537
// MI455X (gfx1250) — compile-verified
//
#include <hip/hip_runtime.h>
#include <cstdint>

// ---------------------------------------------------------------------------
// Problem constants (from the reference)
// ---------------------------------------------------------------------------
#define BB   4
#define TT   16
#define NN   512
#define FF   5
#define HH   64
#define RR   824
#define SS   (BB * NN)     // 2048 sequences
#define GG   (4 * HH)      // 256 gate outputs

typedef __attribute__((ext_vector_type(2))) float v2f;
typedef __attribute__((ext_vector_type(8))) float v8f;

__device__ __forceinline__ float leakyf(float v) { return v > 0.f ? v : 0.01f * v; }
__device__ __forceinline__ float sigmoidf(float v) { return 1.f / (1.f + __expf(-v)); }

// Full-precision CDNA5 matrix core op: D = A(16x4 f32) * B(4x16 f32) + C(16x16 f32)
__device__ __forceinline__ v8f wmma_f32(v2f a, v2f b, v8f c) {
    return __builtin_amdgcn_wmma_f32_16x16x4_f32(
        /*neg_a=*/false, a, /*neg_b=*/false, b,
        /*c_mod=*/(short)0, c, /*reuse_a=*/false, /*reuse_b=*/false);
}

// ---------------------------------------------------------------------------
// Kernel 1: forward LSTM, 16 sequences per block, WMMA recurrent GEMM.
//   gates[16,256] = h[16,64] @ Whh^T[64,256]  (WMMA, K stepped by 4)
//                 + x_t[16,5] @ Wih^T[5,256] + (bih+bhh)   (fused epilogue)
//   wave w owns gate group w: 0=i, 1=f, 2=g, 3=o  (64 cols = 4 N-tiles)
// ---------------------------------------------------------------------------
__global__ __launch_bounds__(128) void lstm_fwd_kernel(
    const float* __restrict__ x,    // [B,T,N,F]
    const float* __restrict__ Wih,  // [256,5]
    const float* __restrict__ Whh,  // [256,64]
    const float* __restrict__ bih,  // [256]
    const float* __restrict__ bhh,  // [256]
    float* __restrict__ hf)         // [S,64]
{
    __shared__ float sh[16][HH];    // current hidden state
    __shared__ float sc[16][HH];    // current cell state
    __shared__ float sg[16][GG];    // gates after nonlinearity

    const int tid  = threadIdx.x;
    const int lane = tid & 31;
    const int wave = tid >> 5;
    const int s0   = blockIdx.x * 16;   // first sequence of this tile

    const int mrow  = lane & 15;        // A row / B col index
    const int khalf = lane >> 4;        // 0 or 1 (which K pair)

    for (int i = tid; i < 16 * HH; i += 128) {
        (&sh[0][0])[i] = 0.f;
        (&sc[0][0])[i] = 0.f;
    }
    __syncthreads();

    for (int t = 0; t < TT; ++t) {
        // ---- recurrent GEMM: 4 N-tiles per wave, K=64 in steps of 4 ----
        v8f acc0 = {}, acc1 = {}, acc2 = {}, acc3 = {};
        for (int k0 = 0; k0 < HH; k0 += 4) {
            const int kk = k0 + 2 * khalf;
            v2f a;
            a.x = sh[mrow][kk];
            a.y = sh[mrow][kk + 1];
            // B[k][n] = Whh^T[k][n] = Whh[gcol][k] (contiguous in k)
            const int gbase = wave * 64 + mrow;
            v2f b0, b1, b2, b3;
            b0.x = Whh[(gbase +  0) * HH + kk]; b0.y = Whh[(gbase +  0) * HH + kk + 1];
            b1.x = Whh[(gbase + 16) * HH + kk]; b1.y = Whh[(gbase + 16) * HH + kk + 1];
            b2.x = Whh[(gbase + 32) * HH + kk]; b2.y = Whh[(gbase + 32) * HH + kk + 1];
            b3.x = Whh[(gbase + 48) * HH + kk]; b3.y = Whh[(gbase + 48) * HH + kk + 1];
            acc0 = wmma_f32(a, b0, acc0);
            acc1 = wmma_f32(a, b1, acc1);
            acc2 = wmma_f32(a, b2, acc2);
            acc3 = wmma_f32(a, b3, acc3);
        }
        // ---- epilogue: + x@Wih^T + bias, nonlinearity, scatter to LDS ----
        v8f accs[4] = {acc0, acc1, acc2, acc3};
        #pragma unroll
        for (int nt = 0; nt < 4; ++nt) {
            const int gcol = wave * 64 + nt * 16 + mrow;
            const float w0 = Wih[gcol * FF + 0];
            const float w1 = Wih[gcol * FF + 1];
            const float w2 = Wih[gcol * FF + 2];
            const float w3 = Wih[gcol * FF + 3];
            const float w4 = Wih[gcol * FF + 4];
            const float bias = bih[gcol] + bhh[gcol];
            #pragma unroll
            for (int r = 0; r < 8; ++r) {
                const int M = r + 8 * khalf;        // D layout: M = r + 8*(lane/16)
                const int s = s0 + M;
                const int bidx = s >> 9;            // s / N
                const int n    = s & (NN - 1);
                const float* xp = x + ((((size_t)bidx * TT + t) * NN) + n) * FF;
                float g = accs[nt][r] + bias
                        + xp[0] * w0 + xp[1] * w1 + xp[2] * w2 + xp[3] * w3 + xp[4] * w4;
                g = (wave == 2) ? tanhf(g) : sigmoidf(g);   // g-gate tanh, others sigmoid
                sg[M][gcol] = g;
            }
        }
        __syncthreads();
        // ---- elementwise state update: c = f*c + i*g ; h = o*tanh(c) ----
        {
            const int row = tid >> 3;           // 0..15
            const int c0  = (tid & 7) * 8;      // 8 h-units per thread
            #pragma unroll
            for (int j = 0; j < 8; ++j) {
                const int col = c0 + j;
                const float iv = sg[row][col];
                const float fv = sg[row][HH + col];
                const float gv = sg[row][2 * HH + col];
                const float ov = sg[row][3 * HH + col];
                const float c  = fv * sc[row][col] + iv * gv;
                sc[row][col] = c;
                sh[row][col] = ov * tanhf(c);
            }
        }
        __syncthreads();
    }

    for (int i = tid; i < 16 * HH; i += 128) {
        const int r = i >> 6, c = i & (HH - 1);
        hf[(size_t)(s0 + r) * HH + c] = sh[r][c];
    }
}

// ---------------------------------------------------------------------------
// Kernel 2: "backward LSTM" = single step on x[:, T-1, :] with zero state.
//   (f-gate and Whh_b vanish because c0 = h0 = 0.)
// ---------------------------------------------------------------------------
__global__ __launch_bounds__(256) void lstm_bwd_kernel(
    const float* __restrict__ x,    // [B,T,N,F]
    const float* __restrict__ Wih,  // [256,5]
    const float* __restrict__ bih,
    const float* __restrict__ bhh,
    float* __restrict__ hb)         // [S,64]
{
    const int idx = blockIdx.x * blockDim.x + threadIdx.x;
    if (idx >= SS * HH) return;
    const int s = idx >> 6, j = idx & (HH - 1);
    const int b = s >> 9, n = s & (NN - 1);
    const float* xp = x + ((((size_t)b * TT + (TT - 1)) * NN) + n) * FF;
    float xv0 = xp[0], xv1 = xp[1], xv2 = xp[2], xv3 = xp[3], xv4 = xp[4];

    auto gate = [&](int g) {
        const float* wr = Wih + g * FF;
        return wr[0] * xv0 + wr[1] * xv1 + wr[2] * xv2 + wr[3] * xv3 + wr[4] * xv4
               + bih[g] + bhh[g];
    };
    const float iv = sigmoidf(gate(j));
    const float gv = tanhf(gate(2 * HH + j));
    const float ov = sigmoidf(gate(3 * HH + j));
    const float c  = iv * gv;
    hb[idx] = ov * tanhf(c);
}

// ---------------------------------------------------------------------------
// Kernel 3: fc0 over [hf | hb] -> output[s][64]
// ---------------------------------------------------------------------------
__global__ __launch_bounds__(256) void fc0_kernel(
    const float* __restrict__ hf, const float* __restrict__ hb,
    const float* __restrict__ w,    // [64,128]
    const float* __restrict__ bvec, // [64]
    float* __restrict__ outp)       // [S,64]
{
    const int idx = blockIdx.x * blockDim.x + threadIdx.x;
    if (idx >= SS * HH) return;
    const int s = idx >> 6, j = idx & (HH - 1);
    const float* wr = w + j * (2 * HH);
    float sum = bvec[j];
    #pragma unroll 8
    for (int k = 0; k < HH; ++k) sum += wr[k] * hf[(size_t)s * HH + k];
    #pragma unroll 8
    for (int k = 0; k < HH; ++k) sum += wr[HH + k] * hb[(size_t)s * HH + k];
    outp[idx] = leakyf(sum);
}

// ---------------------------------------------------------------------------
// Kernel 4: head[s], tail[s] (fc2 / fc3, out-dim 1)
// ---------------------------------------------------------------------------
__global__ __launch_bounds__(256) void headtail_kernel(
    const float* __restrict__ outp,
    const float* __restrict__ w2, const float* __restrict__ b2,
    const float* __restrict__ w3, const float* __restrict__ b3,
    float* __restrict__ head, float* __restrict__ tail)
{
    const int s = blockIdx.x * blockDim.x + threadIdx.x;
    if (s >= SS) return;
    float hsum = 0.f, tsum = 0.f;
    #pragma unroll 8
    for (int k = 0; k < HH; ++k) {
        const float o = outp[(size_t)s * HH + k];
        hsum += o * w2[k];
        tsum += o * w3[k];
    }
    head[s] = leakyf(hsum + b2[0]);
    tail[s] = leakyf(tsum + b3[0]);
}

// ---------------------------------------------------------------------------
// Kernel 5 (roofline path): rel_weight GEMV over 864 MB of rel_encoding.
//   One wave per (n,m) row of 824 floats; float4 streaming + shuffle reduce.
// ---------------------------------------------------------------------------
__global__ __launch_bounds__(256) void relw_kernel(
    const float* __restrict__ rel,  // [N*N, 824]
    const float* __restrict__ w,    // [824]
    const float* __restrict__ bptr, // [1]
    float* __restrict__ rw)         // [N*N]
{
    __shared__ __align__(16) float sw[RR];
    for (int i = threadIdx.x; i < RR; i += 256) sw[i] = w[i];
    __syncthreads();

    const int wave = threadIdx.x >> 5, lane = threadIdx.x & 31;
    const size_t row = (size_t)blockIdx.x * 8 + wave;   // 8 waves per block
    const float4* rp = (const float4*)(rel + row * RR); // RR*4 bytes = 206 float4, 16B aligned
    const float4* wp = (const float4*)sw;

    float sum = 0.f;
    for (int i = lane; i < RR / 4; i += 32) {
        const float4 v  = rp[i];
        const float4 wv = wp[i];
        sum += v.x * wv.x + v.y * wv.y + v.z * wv.z + v.w * wv.w;
    }
    #pragma unroll
    for (int off = 16; off > 0; off >>= 1) sum += __shfl_down(sum, off, 32);
    if (lane == 0) rw[row] = leakyf(sum + bptr[0]);
}

// ---------------------------------------------------------------------------
// Kernel 6: masked softmax over m for each (b,n) row.
//   wm[b,n,m] = softmax_m( rel_mask[n,m] + head[b,n] + tail[b,m] + rw[n,m] )
// ---------------------------------------------------------------------------
__global__ __launch_bounds__(256) void softmax_kernel(
    const float* __restrict__ rw,       // [N,N]
    const float* __restrict__ relmask,  // [N,N]
    const float* __restrict__ head,     // [S]
    const float* __restrict__ tail,     // [S]
    float* __restrict__ wm)             // [B,N,N]
{
    __shared__ float red[256];
    const int bn = blockIdx.x;          // 0..S-1
    const int b = bn >> 9, n = bn & (NN - 1);
    const int tid = threadIdx.x;
    const float hv = head[bn];

    const int m0 = tid, m1 = tid + 256;
    const float e0 = relmask[n * NN + m0] + rw[n * NN + m0] + hv + tail[b * NN + m0];
    const float e1 = relmask[n * NN + m1] + rw[n * NN + m1] + hv + tail[b * NN + m1];

    red[tid] = fmaxf(e0, e1);
    __syncthreads();
    for (int s = 128; s > 0; s >>= 1) {
        if (tid < s) red[tid] = fmaxf(red[tid], red[tid + s]);
        __syncthreads();
    }
    const float mx = red[0];
    __syncthreads();

    const float x0 = __expf(e0 - mx);
    const float x1 = __expf(e1 - mx);
    red[tid] = x0 + x1;
    __syncthreads();
    for (int s = 128; s > 0; s >>= 1) {
        if (tid < s) red[tid] += red[tid + s];
        __syncthreads();
    }
    const float inv = 1.f / red[0];

    float* wrow = wm + ((size_t)b * NN + n) * NN;
    wrow[m0] = x0 * inv;
    wrow[m1] = x1 * inv;
}

// ---------------------------------------------------------------------------
// Kernel 7: graph propagation GEMM with WMMA.
//   proped[b][n][h] = sum_m wm[b][n][m] * output[b][m][h]
//   per block: one (b, 16-row M-tile); wave w owns h-cols [16w,16w+16); K=512.
// ---------------------------------------------------------------------------
__global__ __launch_bounds__(128) void prop_gemm_kernel(
    const float* __restrict__ wm,     // [B,512,512]
    const float* __restrict__ outp,   // [B,512,64]
    float* __restrict__ proped)       // [B,512,64]
{
    const int lane = threadIdx.x & 31;
    const int wave = threadIdx.x >> 5;
    const int b  = blockIdx.x >> 5;
    const int m0 = (blockIdx.x & 31) * 16;
    const int mrow  = lane & 15;
    const int khalf = lane >> 4;

    const float* wmb  = wm   + (size_t)b * NN * NN;
    const float* outb = outp + (size_t)b * NN * HH;
    const int ncol = wave * 16 + mrow;              // h column for B frag

    v8f acc = {};
    const float* arow = wmb + (size_t)(m0 + mrow) * NN;
    for (int k0 = 0; k0 < NN; k0 += 4) {
        const int kk = k0 + 2 * khalf;
        v2f a;  a.x = arow[kk];               a.y = arow[kk + 1];
        v2f bf; bf.x = outb[(size_t)kk * HH + ncol];
                bf.y = outb[(size_t)(kk + 1) * HH + ncol];
        acc = wmma_f32(a, bf, acc);
    }
    #pragma unroll
    for (int r = 0; r < 8; ++r) {
        const int row = m0 + r + 8 * khalf;   // D layout: M = r + 8*(lane/16)
        proped[((size_t)b * NN + row) * HH + wave * 16 + mrow] = acc[r];
    }
}

// ---------------------------------------------------------------------------
// Kernel 8: final prediction: leaky([output | proped] @ pred_w^T + pred_b)
// ---------------------------------------------------------------------------
__global__ __launch_bounds__(256) void pred_kernel(
    const float* __restrict__ outp, const float* __restrict__ proped,
    const float* __restrict__ pw,   // [128]
    const float* __restrict__ pb,   // [1]
    float* __restrict__ out)        // [S]
{
    const int s = blockIdx.x * blockDim.x + threadIdx.x;
    if (s >= SS) return;
    float sum = pb[0];
    #pragma unroll 8
    for (int k = 0; k < HH; ++k) sum += pw[k]      * outp[(size_t)s * HH + k];
    #pragma unroll 8
    for (int k = 0; k < HH; ++k) sum += pw[HH + k] * proped[(size_t)s * HH + k];
    out[s] = leakyf(sum);
}

// ---------------------------------------------------------------------------
// Host launcher
// ---------------------------------------------------------------------------
extern "C" void kernel_launch(void* const* d_in, const int* in_sizes, int n_in,
                              void* d_out, int out_size, void* d_ws, size_t ws_size,
                              hipStream_t stream) {
    const float* x       = (const float*)d_in[0];
    const float* rel     = (const float*)d_in[1];
    const float* relmask = (const float*)d_in[2];
    const float* Wih_f   = (const float*)d_in[3];
    const float* Whh_f   = (const float*)d_in[4];
    const float* bih_f   = (const float*)d_in[5];
    const float* bhh_f   = (const float*)d_in[6];
    const float* Wih_b   = (const float*)d_in[7];
    // d_in[8] = Whh_b: provably unused (h0 == 0 for the single used backward step)
    const float* bih_b   = (const float*)d_in[9];
    const float* bhh_b   = (const float*)d_in[10];
    const float* fc0_w   = (const float*)d_in[11];
    const float* fc0_b   = (const float*)d_in[12];
    const float* fc1_w   = (const float*)d_in[13];
    const float* fc1_b   = (const float*)d_in[14];
    const float* fc2_w   = (const float*)d_in[15];
    const float* fc2_b   = (const float*)d_in[16];
    const float* fc3_w   = (const float*)d_in[17];
    const float* fc3_b   = (const float*)d_in[18];
    const float* pred_w  = (const float*)d_in[19];
    const float* pred_b  = (const float*)d_in[20];
    // d_in[21] = all_one: broadcast only, folded into the weight construction

    float* ws = (float*)d_ws;
    float* hf     = ws;                       // S*H      = 131072
    float* hb     = hf     + SS * HH;         // S*H      = 131072
    float* outp   = hb     + SS * HH;         // S*H      = 131072
    float* head   = outp   + SS * HH;         // S        = 2048
    float* tail   = head   + SS;              // S        = 2048
    float* rw     = tail   + SS;              // N*N      = 262144
    float* wm     = rw     + NN * NN;         // B*N*N    = 1048576
    float* proped = wm     + (size_t)BB * NN * NN;  // S*H = 131072

    // 1) forward LSTM (WMMA recurrent GEMM)
    lstm_fwd_kernel<<<SS / 16, 128, 0, stream>>>(x, Wih_f, Whh_f, bih_f, bhh_f, hf);
    // 2) backward single step
    lstm_bwd_kernel<<<(SS * HH + 255) / 256, 256, 0, stream>>>(x, Wih_b, bih_b, bhh_b, hb);
    // 3) fc0
    fc0_kernel<<<(SS * HH + 255) / 256, 256, 0, stream>>>(hf, hb, fc0_w, fc0_b, outp);
    // 4) head / tail
    headtail_kernel<<<(SS + 255) / 256, 256, 0, stream>>>(outp, fc2_w, fc2_b, fc3_w, fc3_b,
                                                          head, tail);
    // 5) rel_weight GEMV (bandwidth roofline: 864 MB stream)
    relw_kernel<<<(NN * NN) / 8, 256, 0, stream>>>(rel, fc1_w, fc1_b, rw);
    // 6) masked softmax
    softmax_kernel<<<SS, 256, 0, stream>>>(rw, relmask, head, tail, wm);
    // 7) propagation GEMM (WMMA)
    prop_gemm_kernel<<<BB * (NN / 16), 128, 0, stream>>>(wm, outp, proped);
    // 8) prediction
    pred_kernel<<<(SS + 255) / 256, 256, 0, stream>>>(outp, proped, pred_w, pred_b,
                                                      (float*)d_out);
}